// PredictionDecoder_83983790506169
// MI455X (gfx1250) — compile-verified
//
#include <hip/hip_runtime.h>
#include <hip/hip_bf16.h>
#include <math.h>

// ---------------- problem constants (from reference) ----------------
#define B_SZ     32
#define N_ANCH   49104
#define N_CLS    3
#define K_PRE    200
#define K_PAD    256          // bitonic sort width
#define SCORE_TH 0.05f
#define IOU_THR  0.5f
#define PT       208          // padded 200 -> 13 * 16 for WMMA tiles
#define NTILE    13

// ---------------- CDNA5 feature probes ----------------
#if defined(__gfx1250__) && __has_builtin(__builtin_amdgcn_global_load_async_to_lds_b32)
#define HAVE_ASYNC 1
#else
#define HAVE_ASYNC 0
#endif

#if defined(__gfx1250__) && __has_builtin(__builtin_amdgcn_wmma_f32_16x16x4_f32)
#define HAVE_WMMA4 1
#else
#define HAVE_WMMA4 0
#endif

typedef float v2f __attribute__((ext_vector_type(2)));
typedef float v8f __attribute__((ext_vector_type(8)));

#if HAVE_ASYNC
typedef __attribute__((address_space(1))) int* gas_i32p;
typedef __attribute__((address_space(3))) int* las_i32p;
#endif

// async copy one dword global -> LDS (CDNA5 ASYNCcnt path)
__device__ __forceinline__ void async_g2l_b32(unsigned* ldsDst, const float* gSrc) {
#if HAVE_ASYNC
  __builtin_amdgcn_global_load_async_to_lds_b32(
      (gas_i32p)gSrc, (las_i32p)ldsDst, 0, 0);
#else
  *ldsDst = __float_as_uint(*gSrc);
#endif
}

__device__ __forceinline__ void wait_async0() {
#if HAVE_ASYNC
#if __has_builtin(__builtin_amdgcn_s_wait_asynccnt)
  __builtin_amdgcn_s_wait_asynccnt(0);
#else
  asm volatile("s_wait_asynccnt 0" ::: "memory");
#endif
#endif
}

// monotonic float -> uint key (descending top-k == largest keys)
__device__ __forceinline__ unsigned f2key(unsigned u) {
  return (u & 0x80000000u) ? ~u : (u | 0x80000000u);
}
__device__ __forceinline__ float key2f(unsigned k) {
  unsigned u = (k & 0x80000000u) ? (k & 0x7fffffffu) : ~k;
  return __uint_as_float(u);
}

// anchor generator (image_size = 512): replicates make_anchors arithmetically
__device__ __forceinline__ void anchor_for(int n, float& cx, float& cy,
                                           float& aw, float& ah) {
  const int fss[5]     = {64, 32, 16, 8, 4};
  const int strides[5] = {8, 16, 32, 64, 128};
  int off = 0, lvl = 0, fs = 64;
  for (int i = 0; i < 5; ++i) {
    int cnt = fss[i] * fss[i] * 9;
    if (n < off + cnt) { lvl = i; fs = fss[i]; break; }
    off += cnt;
  }
  int rem  = n - off;
  int k    = rem % 9;
  int cell = rem / 9;
  int j  = cell % fs;       // x index (meshgrid 'xy')
  int i2 = cell / fs;       // y index
  float stride = (float)strides[lvl];
  cx = (j + 0.5f) * stride;
  cy = (i2 + 0.5f) * stride;
  float side = (float)(lvl * 120 + 32);   // step = (512-32)//4 = 120
  float area = side * side;
  int ri = k / 3;
  float r = (ri == 0) ? 0.5f : ((ri == 1) ? 1.0f : 2.0f);
  float s = exp2f((float)(k % 3) * (1.0f / 3.0f));
  float h = sqrtf(area / r);
  float w = area / h;
  aw = s * w;
  ah = s * h;
}

// =====================================================================
// Kernel A: per (b,c) block. Async-stage score channel to LDS once,
// radix-select top-200 logits, bitonic-sort (desc, index tiebreak),
// decode boxes for survivors, write to workspace.
// =====================================================================
__global__ __launch_bounds__(256) void topk_decode_kernel(
    const float* __restrict__ preds,
    float* __restrict__ ws_boxes,    // [96][200][4] corner boxes
    float* __restrict__ ws_scores) { // [96][200] sigmoid scores (desc)
  extern __shared__ char dynsmem[];
  unsigned* keys = (unsigned*)dynsmem;          // N_ANCH sortable keys

  __shared__ unsigned hist[256];
  __shared__ unsigned sel_key[K_PAD];
  __shared__ int      sel_idx[K_PAD];
  __shared__ unsigned s_b1, s_b2, s_needed, s_needed2, s_cnt, s_q;

  const int tid = threadIdx.x;
  const int blk = blockIdx.x;            // 0..95
  const int b = blk / N_CLS;
  const int c = blk % N_CLS;

  // ---- stage score channel (stride 7 floats) into LDS via async path ----
  const float* sbase = preds + (size_t)b * N_ANCH * 7 + 4 + c;
  for (int n = tid; n < N_ANCH; n += 256)
    async_g2l_b32(&keys[n], sbase + (size_t)n * 7);
  wait_async0();
  __syncthreads();

  // convert in place to sortable keys (each slot owned by one thread)
  for (int n = tid; n < N_ANCH; n += 256) keys[n] = f2key(keys[n]);

  // ---- pass 1: histogram top byte ----
  hist[tid] = 0;
  __syncthreads();
  for (int n = tid; n < N_ANCH; n += 256)
    atomicAdd(&hist[keys[n] >> 24], 1u);
  __syncthreads();
  if (tid == 0) {
    unsigned cum = 0; int bb = 0; unsigned need = K_PRE;
    for (int i = 255; i >= 0; --i) {
      if (cum + hist[i] >= K_PRE) { bb = i; need = K_PRE - cum; break; }
      cum += hist[i];
    }
    s_b1 = (unsigned)bb; s_needed = need;
  }
  __syncthreads();
  const unsigned b1 = s_b1;
  const unsigned need1 = s_needed;

  // ---- pass 2: histogram second byte within bucket b1 ----
  hist[tid] = 0;
  __syncthreads();
  for (int n = tid; n < N_ANCH; n += 256) {
    unsigned k = keys[n];
    if ((k >> 24) == b1) atomicAdd(&hist[(k >> 16) & 0xffu], 1u);
  }
  __syncthreads();
  if (tid == 0) {
    unsigned cum = 0; int bb = 0; unsigned need = need1;
    for (int i = 255; i >= 0; --i) {
      if (cum + hist[i] >= need1) { bb = i; need = need1 - cum; break; }
      cum += hist[i];
    }
    s_b2 = (unsigned)bb; s_needed2 = need;
    s_cnt = 0; s_q = 0;
  }
  sel_key[tid] = 0u;
  sel_idx[tid] = 0x7fffffff;
  __syncthreads();

  // ---- pass 3: compact the top-200 (order fixed by the sort below) ----
  const unsigned Pfull   = (b1 << 8) | s_b2;
  const unsigned needed2 = s_needed2;
  for (int n = tid; n < N_ANCH; n += 256) {
    unsigned k = keys[n];
    unsigned p = k >> 16;
    bool take = false;
    if (p > Pfull) take = true;
    else if (p == Pfull) take = (atomicAdd(&s_q, 1u) < needed2);
    if (take) {
      unsigned s = atomicAdd(&s_cnt, 1u);
      if (s < K_PAD) { sel_key[s] = k; sel_idx[s] = n; }
    }
  }
  __syncthreads();

  // ---- bitonic sort 256 wide: key desc, index asc on ties (== lax.top_k) ----
  for (int k = 2; k <= K_PAD; k <<= 1) {
    for (int j = k >> 1; j > 0; j >>= 1) {
      __syncthreads();
      int ixj = tid ^ j;
      if (ixj > tid) {
        unsigned ka = sel_key[tid], kb = sel_key[ixj];
        int ia = sel_idx[tid], ib = sel_idx[ixj];
        bool a_better = (ka > kb) || (ka == kb && ia < ib);
        bool descBlk = ((tid & k) == 0);
        bool doswap = descBlk ? (!a_better) : a_better;
        if (doswap) {
          sel_key[tid] = kb; sel_key[ixj] = ka;
          sel_idx[tid] = ib; sel_idx[ixj] = ia;
        }
      }
    }
  }
  __syncthreads();

  // ---- decode boxes for the 200 survivors, write sorted to workspace ----
  if (tid < K_PRE) {
    int n = sel_idx[tid];
    float logit = key2f(sel_key[tid]);
    float score = 1.0f / (1.0f + expf(-logit));
    if (n >= N_ANCH || n < 0) { n = 0; score = -1.0f; }  // defensive

    float cx, cy, aw, ah;
    anchor_for(n, cx, cy, aw, ah);
    const float* bp = preds + ((size_t)b * N_ANCH + n) * 7;
    float tx = bp[0] * 0.1f, ty = bp[1] * 0.1f;
    float tw = bp[2] * 0.2f, th = bp[3] * 0.2f;
    float x = tx * aw + cx, y = ty * ah + cy;
    float w = expf(tw) * aw, h = expf(th) * ah;

    size_t base = ((size_t)blk * K_PRE + tid);
    ws_boxes[base * 4 + 0] = x - 0.5f * w;
    ws_boxes[base * 4 + 1] = y - 0.5f * h;
    ws_boxes[base * 4 + 2] = x + 0.5f * w;
    ws_boxes[base * 4 + 3] = y + 0.5f * h;
    ws_scores[base] = score;
  }
}

// =====================================================================
// Kernel B: per batch block. For each class: full 200x200 IoU matrix
// (WMMA f32 16x16x4 computes the area_i + area_j union base per tile,
// VALU does intersections), greedy NMS (== reference fori_loop), take
// first 10 kept; then merge 3x10 -> top-10, write outputs.
// =====================================================================
__global__ __launch_bounds__(256) void nms_merge_kernel(
    const float* __restrict__ ws_boxes,
    const float* __restrict__ ws_scores,
    float* __restrict__ dout) {
  extern __shared__ char dynsmem[];
  float* iou = (float*)dynsmem;                // [PT * PT]

  __shared__ float bx[PT][4];
  __shared__ float sarea[PT];
  __shared__ float ss[PT];
  __shared__ int   keep[K_PRE];
  __shared__ float out_s[32];                  // 30 used: [class][10]
  __shared__ float out_b[32][4];
  __shared__ float out_c[32];

  const int tid = threadIdx.x;
  const int b = blockIdx.x;
  const int wave = tid >> 5;
  const int lane = tid & 31;

  for (int c = 0; c < N_CLS; ++c) {
    __syncthreads();
    const int bcls = b * N_CLS + c;

    // load sorted candidates, pad to PT
    for (int i = tid; i < PT; i += 256) {
      if (i < K_PRE) {
        size_t base = ((size_t)bcls * K_PRE + i);
        float x1 = ws_boxes[base * 4 + 0];
        float y1 = ws_boxes[base * 4 + 1];
        float x2 = ws_boxes[base * 4 + 2];
        float y2 = ws_boxes[base * 4 + 3];
        bx[i][0] = x1; bx[i][1] = y1; bx[i][2] = x2; bx[i][3] = y2;
        sarea[i] = (x2 - x1) * (y2 - y1);
        ss[i] = ws_scores[base];
      } else {
        bx[i][0] = bx[i][1] = bx[i][2] = bx[i][3] = 0.0f;
        sarea[i] = 0.0f;
        ss[i] = -1.0f;
      }
    }
    __syncthreads();

    // ---- IoU tiles: 13x13 tiles of 16x16, one wave per tile ----
    for (int t = wave; t < NTILE * NTILE; t += 8) {
      int ti = t / NTILE, tj = t % NTILE;
      int rb = ti * 16, cb = tj * 16;

      float ubase[8];
      int M0 = (lane < 16) ? 0 : 8;
      int N  = lane & 15;
      int ccol = cb + N;

#if HAVE_WMMA4
      // A(16x4): row M -> [area_row[M], 1, 0, 0]; B(4x16): col N -> [1, area_col[N], 0, 0]^T
      float arl = (lane < 16) ? sarea[rb + N] : 0.0f;
      float acl = (lane < 16) ? sarea[cb + N] : 0.0f;
      float one = (lane < 16) ? 1.0f : 0.0f;
      v2f av;  av.x = arl; av.y = one;
      v2f bv;  bv.x = one; bv.y = acl;
      v8f cz = {};
      v8f d = __builtin_amdgcn_wmma_f32_16x16x4_f32(
          false, av, false, bv, (short)0, cz, false, false);
      #pragma unroll
      for (int v = 0; v < 8; ++v) ubase[v] = d[v];
#else
      #pragma unroll
      for (int v = 0; v < 8; ++v) ubase[v] = sarea[rb + M0 + v] + sarea[ccol];
#endif

      float cx1 = bx[ccol][0], cy1 = bx[ccol][1];
      float cx2 = bx[ccol][2], cy2 = bx[ccol][3];
      #pragma unroll
      for (int v = 0; v < 8; ++v) {
        int r = rb + M0 + v;
        float ix = fminf(bx[r][2], cx2) - fmaxf(bx[r][0], cx1);
        float iy = fminf(bx[r][3], cy2) - fmaxf(bx[r][1], cy1);
        ix = fmaxf(ix, 0.0f);
        iy = fmaxf(iy, 0.0f);
        float inter = ix * iy;
        float uni = ubase[v] - inter;
        iou[r * PT + ccol] = inter / fmaxf(uni, 1e-8f);
      }
    }

    // ---- greedy NMS in score order (matches reference fori_loop) ----
    for (int i = tid; i < K_PRE; i += 256) keep[i] = (ss[i] >= SCORE_TH) ? 1 : 0;
    __syncthreads();
    for (int i = 0; i < K_PRE; ++i) {
      if (keep[i]) {
        for (int j = i + 1 + tid; j < K_PRE; j += 256)
          if (iou[i * PT + j] > IOU_THR) keep[j] = 0;
      }
      __syncthreads();
    }

    // ---- first up-to-10 kept (already score-descending) ----
    if (tid == 0) {
      int cnt = 0;
      for (int i = 0; i < K_PRE && cnt < 10; ++i) {
        if (keep[i]) {
          int slot = c * 10 + cnt;
          out_s[slot] = ss[i];
          out_b[slot][0] = bx[i][0]; out_b[slot][1] = bx[i][1];
          out_b[slot][2] = bx[i][2]; out_b[slot][3] = bx[i][3];
          out_c[slot] = (float)c;
          ++cnt;
        }
      }
      for (; cnt < 10; ++cnt) {
        int slot = c * 10 + cnt;
        out_s[slot] = -1.0f;
        out_b[slot][0] = out_b[slot][1] = out_b[slot][2] = out_b[slot][3] = 0.0f;
        out_c[slot] = (float)c;
      }
    }
    __syncthreads();
  }

  // ---- final merge 30 -> 10 (ties -> lower flat index, like top_k) ----
  if (tid == 0) {
    bool used[30];
    for (int m = 0; m < 30; ++m) used[m] = false;
    int valid = 0;
    for (int k = 0; k < 10; ++k) {
      int best = 0; float bs = -1e30f;
      for (int m = 0; m < 30; ++m)
        if (!used[m] && out_s[m] > bs) { bs = out_s[m]; best = m; }
      used[best] = true;
      bool vm = bs > 0.0f;
      float* ob = dout + ((size_t)b * 10 + k) * 4;
      ob[0] = vm ? out_b[best][0] : 0.0f;
      ob[1] = vm ? out_b[best][1] : 0.0f;
      ob[2] = vm ? out_b[best][2] : 0.0f;
      ob[3] = vm ? out_b[best][3] : 0.0f;
      dout[B_SZ * 10 * 4 + (size_t)b * 10 + k]            = vm ? bs : 0.0f;
      dout[B_SZ * 10 * 4 + B_SZ * 10 + (size_t)b * 10 + k] = vm ? out_c[best] : 0.0f;
      if (vm) ++valid;
    }
    // valid_detections is int32 in the reference tuple: store raw i32 bits
    ((int*)dout)[B_SZ * 10 * 4 + 2 * B_SZ * 10 + b] = valid;
  }
}

// =====================================================================
extern "C" void kernel_launch(void* const* d_in, const int* in_sizes, int n_in,
                              void* d_out, int out_size, void* d_ws, size_t ws_size,
                              hipStream_t stream) {
  (void)in_sizes; (void)n_in; (void)out_size; (void)ws_size;
  // inputs: d_in[0] = images (unused, only shape matters), d_in[1] = predictions
  const float* preds = (const float*)d_in[1];

  float* ws_boxes  = (float*)d_ws;                                  // 96*200*4 f32
  float* ws_scores = ws_boxes + (size_t)B_SZ * N_CLS * K_PRE * 4;   // 96*200   f32

  // Kernel A: 96 blocks, dynamic LDS holds the 49104 score keys (192 KB < 320 KB/WGP)
  size_t ldsA = (size_t)((N_ANCH + 255) & ~255) * sizeof(unsigned);
  topk_decode_kernel<<<B_SZ * N_CLS, 256, ldsA, stream>>>(preds, ws_boxes, ws_scores);

  // Kernel B: 32 blocks, dynamic LDS holds the 208x208 IoU matrix (169 KB)
  size_t ldsB = (size_t)PT * PT * sizeof(float);
  nms_merge_kernel<<<B_SZ, 256, ldsB, stream>>>(ws_boxes, ws_scores, (float*)d_out);
}